// MultiHeadAttentionBlock_16363825398541
// MI455X (gfx1250) — compile-verified
//
#include <hip/hip_runtime.h>
#include <hip/hip_bf16.h>
#include <math.h>

#define HIDDEN  4096
#define HEADS   32
#define DK      128
#define BS      16
#define MAXSEQ  2048
#define KSPLIT  16
#define KCHUNK  (HIDDEN / KSPLIT)   // 256

typedef float v2f __attribute__((ext_vector_type(2)));
typedef float v8f __attribute__((ext_vector_type(8)));

// ---------------------------------------------------------------------------
// Partial GEMM: P[w][ks][0:16][col0:col0+16] = X[0:16, kbeg:kbeg+256] @ W[kbeg:kbeg+256, col0:col0+16]
// One wave per (tile, ks) pair: 256 tiles x 16 k-splits = 4096 waves per
// weight; blockIdx.y selects among up to 3 weights (QKV fused in one launch
// -> 12288 waves, enough parallelism to reach the HBM roofline).
// fp32 V_WMMA_F32_16X16X4_F32; M=16 matches the batch dimension exactly.
// ---------------------------------------------------------------------------
__global__ __launch_bounds__(256) void gemm_x16_partial(
    const float* __restrict__ X,
    const float* __restrict__ W0, const float* __restrict__ W1,
    const float* __restrict__ W2, float* __restrict__ P)
{
  const int wave = threadIdx.x >> 5;
  const int lane = threadIdx.x & 31;
  const int gw   = blockIdx.x * 8 + wave;   // 0..4095
  const int tile = gw >> 4;                 // 0..255 column tile
  const int ks   = gw & (KSPLIT - 1);       // 0..15  k-split
  const int col0 = tile * 16;
  const int m    = lane & 15;               // row (A) / column (B) in tile
  const int kh   = (lane >> 4) << 1;        // 0 or 2: which K pair
  const int w    = blockIdx.y;

  const float* W  = (w == 0) ? W0 : (w == 1) ? W1 : W2;
  float*       Pw = P + (size_t)w * KSPLIT * BS * HIDDEN;

  const int    kbeg = ks * KCHUNK;
  const float* xrow = X + (size_t)m * HIDDEN + kbeg;
  const float* wcol = W + (size_t)kbeg * HIDDEN + col0 + m;

  v8f acc = {0.f, 0.f, 0.f, 0.f, 0.f, 0.f, 0.f, 0.f};

  #pragma unroll 8
  for (int kk = 0; kk < KCHUNK; kk += 4) {
    v2f a, b;
    a.x = xrow[kk + kh];
    a.y = xrow[kk + kh + 1];
    b.x = wcol[(size_t)(kk + kh) * HIDDEN];
    b.y = wcol[(size_t)(kk + kh + 1) * HIDDEN];
    acc = __builtin_amdgcn_wmma_f32_16x16x4_f32(
        false, a, false, b, (short)0, acc, false, false);
  }

  const int rbase = (lane >> 4) << 3;       // 0 or 8
  #pragma unroll
  for (int r = 0; r < 8; ++r) {
    Pw[((size_t)ks * BS + rbase + r) * HIDDEN + col0 + m] = acc[r];
  }
}

// ---------------------------------------------------------------------------
// Deterministic k-split reduction: Y = sum_{ks} P[w][ks]  (fixed order).
// float4 per thread, fully coalesced.
// ---------------------------------------------------------------------------
__global__ __launch_bounds__(256) void reduce_ksplit(
    const float* __restrict__ P, float* __restrict__ Y0,
    float* __restrict__ Y1, float* __restrict__ Y2)
{
  const int w = blockIdx.y;
  float* Y = (w == 0) ? Y0 : (w == 1) ? Y1 : Y2;
  const float* Pw = P + (size_t)w * KSPLIT * BS * HIDDEN;

  const size_t i = ((size_t)blockIdx.x * 256 + threadIdx.x) * 4;
  float4 s = *(const float4*)(Pw + i);
  #pragma unroll
  for (int ks = 1; ks < KSPLIT; ++ks) {
    const float4 t = *(const float4*)(Pw + (size_t)ks * BS * HIDDEN + i);
    s.x += t.x; s.y += t.y; s.z += t.z; s.w += t.w;
  }
  *(float4*)(Y + i) = s;
}

// ---------------------------------------------------------------------------
// Attention for one (batch, head) per workgroup. 256 threads = 8 waves.
// Scores: wave-cooperative dot (float4/lane, shuffle reduce). Softmax in LDS.
// Weighted V: lane owns 4 contiguous dims, waves stripe positions.
// Position start_pos reads the freshly projected k/v from workspace (the
// input caches are never written). Position loops unrolled 4x to keep ~2KB
// of b128 loads in flight per wave (HBM roofline needs deep MLP).
// ---------------------------------------------------------------------------
__global__ __launch_bounds__(256) void attn_kernel(
    const float* __restrict__ qb, const float* __restrict__ kn,
    const float* __restrict__ vn, const float* __restrict__ ck,
    const float* __restrict__ cv, const int* __restrict__ spp,
    float* __restrict__ ob)
{
  __shared__ float sc[MAXSEQ];
  __shared__ float pout[8 * DK];
  __shared__ float wred[8];
  __shared__ float wsum[8];

  const int h    = blockIdx.x;
  const int b    = blockIdx.y;
  const int tid  = threadIdx.x;
  const int wave = tid >> 5;
  const int lane = tid & 31;
  const int sp   = *spp;
  const int seq  = sp + 1;
  const float scale = 0.08838834764831845f;    // 1/sqrt(128)

  const float* qp  = qb + ((size_t)b * HEADS + h) * DK;
  const float* knp = kn + ((size_t)b * HEADS + h) * DK;
  const float* vnp = vn + ((size_t)b * HEADS + h) * DK;

  const float4 q4 = *(const float4*)(qp + lane * 4);

  // ---- scores = (q . k) * scale -------------------------------------------
  #pragma unroll 4
  for (int p = wave; p < seq; p += 8) {
    const float* kp = (p == sp)
        ? knp
        : ck + (((size_t)b * MAXSEQ + p) * HEADS + h) * DK;
    const float4 k4 = *(const float4*)(kp + lane * 4);
    float d = q4.x * k4.x + q4.y * k4.y + q4.z * k4.z + q4.w * k4.w;
    #pragma unroll
    for (int off = 16; off > 0; off >>= 1) d += __shfl_xor(d, off, 32);
    if (lane == 0) sc[p] = d * scale;
  }
  __syncthreads();

  // ---- softmax max --------------------------------------------------------
  float lm = -INFINITY;
  for (int p = tid; p < seq; p += 256) lm = fmaxf(lm, sc[p]);
  #pragma unroll
  for (int off = 16; off > 0; off >>= 1) lm = fmaxf(lm, __shfl_xor(lm, off, 32));
  if (lane == 0) wred[wave] = lm;
  __syncthreads();
  float gmax = wred[0];
  #pragma unroll
  for (int i = 1; i < 8; ++i) gmax = fmaxf(gmax, wred[i]);

  // ---- exp + sum ----------------------------------------------------------
  float ls = 0.f;
  for (int p = tid; p < seq; p += 256) {
    float e = __expf(sc[p] - gmax);
    sc[p] = e;
    ls += e;
  }
  #pragma unroll
  for (int off = 16; off > 0; off >>= 1) ls += __shfl_xor(ls, off, 32);
  if (lane == 0) wsum[wave] = ls;
  __syncthreads();
  float gsum = 0.f;
  #pragma unroll
  for (int i = 0; i < 8; ++i) gsum += wsum[i];
  const float inv = 1.0f / gsum;

  // ---- out = attn @ V -----------------------------------------------------
  float4 acc = {0.f, 0.f, 0.f, 0.f};
  #pragma unroll 4
  for (int p = wave; p < seq; p += 8) {
    const float* vp = (p == sp)
        ? vnp
        : cv + (((size_t)b * MAXSEQ + p) * HEADS + h) * DK;
    const float4 v4 = *(const float4*)(vp + lane * 4);
    const float w = sc[p];
    acc.x += w * v4.x; acc.y += w * v4.y;
    acc.z += w * v4.z; acc.w += w * v4.w;
  }
  *(float4*)(pout + wave * DK + lane * 4) = acc;
  __syncthreads();

  if (tid < DK) {
    float s = 0.f;
    #pragma unroll
    for (int w = 0; w < 8; ++w) s += pout[w * DK + tid];
    ob[((size_t)b * HEADS + h) * DK + tid] = s * inv;
  }
}

// ---------------------------------------------------------------------------
extern "C" void kernel_launch(void* const* d_in, const int* in_sizes, int n_in,
                              void* d_out, int out_size, void* d_ws, size_t ws_size,
                              hipStream_t stream)
{
  (void)in_sizes; (void)n_in; (void)out_size; (void)ws_size;

  const float* x  = (const float*)d_in[0];
  const float* ck = (const float*)d_in[1];
  const float* cv = (const float*)d_in[2];
  const float* wq = (const float*)d_in[3];
  const float* wk = (const float*)d_in[4];
  const float* wv = (const float*)d_in[5];
  const float* wo = (const float*)d_in[6];
  const int*   sp = (const int*)d_in[7];
  float* out = (float*)d_out;

  float* qb   = (float*)d_ws;                 // 16*4096 each
  float* kb   = qb + BS * HIDDEN;
  float* vb   = kb + BS * HIDDEN;
  float* ab   = vb + BS * HIDDEN;
  float* part = ab + BS * HIDDEN;             // 3 * 16 * 16 * 4096 floats (12 MB)

  const dim3 block(256);
  const dim3 gemm_grid(512);                  // 4096 waves per weight
  const dim3 red_grid(64);                    // 65536 elems / (256*4)

  // QKV fused: 12288 waves streaming 192 MB of weights once.
  gemm_x16_partial<<<dim3(512, 3), block, 0, stream>>>(x, wq, wk, wv, part);
  reduce_ksplit<<<dim3(64, 3), block, 0, stream>>>(part, qb, kb, vb);

  attn_kernel<<<dim3(HEADS, BS), block, 0, stream>>>(qb, kb, vb, ck, cv, sp, ab);

  // Output projection (re-uses the same partial buffer).
  gemm_x16_partial<<<dim3(512, 1), block, 0, stream>>>(ab, wo, wo, wo, part);
  reduce_ksplit<<<dim3(64, 1), block, 0, stream>>>(part, out, out, out);
}